// GNN_14181982012210
// MI455X (gfx1250) — compile-verified
//
#include <hip/hip_runtime.h>
#include <math.h>

// Problem constants (match reference)
#define NN 10000     // nodes
#define EE 160000    // raw edges
#define ET 170000    // edges + self loops
#define DD 128       // feature dim (= H*C for every layer)
#define LMID 19

typedef __attribute__((ext_vector_type(2))) float v2f;
typedef __attribute__((ext_vector_type(8))) float v8f;

// ---- monotone float <-> uint ordering (for exact atomic max on floats) ----
__device__ __forceinline__ unsigned f2ord(float f) {
  unsigned u = __float_as_uint(f);
  return (u & 0x80000000u) ? ~u : (u | 0x80000000u);
}
__device__ __forceinline__ float ord2f(unsigned u) {
  return (u & 0x80000000u) ? __uint_as_float(u & 0x7fffffffu)
                           : __uint_as_float(~u);
}

// ---------------------------------------------------------------------------
// h = X @ W  using V_WMMA_F32_16X16X4_F32 (full f32 precision).
// One wave computes one 16x16 tile of h. Block = 8 waves = 16 rows x 128 cols.
// grid.x = 625 row-tiles (10000 = 625*16 exactly).
//
// A (16x4 f32) lane layout: lanes 0-15 rows 0-15 {K=0,K=1}; lanes 16-31 {K=2,K=3}.
// B (4x16 f32) lane layout: v0 = {K=0 | K=2} x N=lane&15 ; v1 = {K=1 | K=3}.
// D (16x16 f32): VGPR r -> row r (lanes 0-15) / row r+8 (lanes 16-31), col = lane&15.
// ---------------------------------------------------------------------------
__global__ __launch_bounds__(256) void gat_gemm(const float* __restrict__ X,
                                                const float* __restrict__ W,
                                                float* __restrict__ Hout) {
  const int lane  = threadIdx.x & 31;
  const int wave  = threadIdx.x >> 5;
  const int row0  = blockIdx.x * 16;
  const int col0  = wave * 16;
  const int rsel  = lane & 15;
  const int khalf = lane >> 4;              // 0 or 1

  v8f acc = {};
  const float* Xrow = X + (row0 + rsel) * DD;
#pragma unroll 4
  for (int k = 0; k < DD; k += 4) {
    v2f a, b;
    const float* ap = Xrow + k + (khalf << 1);
    a.x = ap[0];
    a.y = ap[1];
    const float* bp = W + (k + (khalf << 1)) * DD + col0 + rsel;
    b.x = bp[0];
    b.y = bp[DD];
    acc = __builtin_amdgcn_wmma_f32_16x16x4_f32(
        /*neg_a=*/false, a, /*neg_b=*/false, b,
        /*c_mod=*/(short)0, acc, /*reuse_a=*/false, /*reuse_b=*/false);
  }

  float* out = Hout + (row0 + (khalf << 3)) * DD + col0 + rsel;
#pragma unroll
  for (int r = 0; r < 8; ++r) out[r * DD] = acc[r];
}

// ---------------------------------------------------------------------------
// es[n,h] = <h[n,h,:], a_src[h,:]> ; ed likewise. One thread per (n, head).
// ---------------------------------------------------------------------------
template <int HH, int CC>
__global__ __launch_bounds__(256) void gat_attn(const float* __restrict__ Hf,
                                                const float* __restrict__ asrc,
                                                const float* __restrict__ adst,
                                                float* __restrict__ es,
                                                float* __restrict__ ed) {
  int idx = blockIdx.x * blockDim.x + threadIdx.x;
  if (idx >= NN * HH) return;
  int n = idx / HH, hd = idx - n * HH;
  const float* hp = Hf + n * DD + hd * CC;
  const float* as = asrc + hd * CC;
  const float* ad = adst + hd * CC;
  float s = 0.f, d = 0.f;
#pragma unroll 8
  for (int c = 0; c < CC; ++c) {
    float v = hp[c];
    s = fmaf(v, as[c], s);
    d = fmaf(v, ad[c], d);
  }
  es[idx] = s;
  ed[idx] = d;
}

// Zero accumulators: agg[N*DD] = 0, m[N*HH] = 0 (== ordered -inf), z[N*HH] = 0.
__global__ __launch_bounds__(256) void gat_init(unsigned* __restrict__ m,
                                                float* __restrict__ z,
                                                float* __restrict__ agg,
                                                int nh) {
  int idx = blockIdx.x * blockDim.x + threadIdx.x;
  if (idx < NN * DD) agg[idx] = 0.f;
  if (idx < nh) { m[idx] = 0u; z[idx] = 0.f; }
}

// Pass 1: e = leaky_relu(es[src]+ed[dst]); segment-max into m[dst].
template <int HH>
__global__ __launch_bounds__(256) void gat_edge_max(const int* __restrict__ ei,
                                                    const float* __restrict__ es,
                                                    const float* __restrict__ ed,
                                                    float* __restrict__ e_buf,
                                                    unsigned* __restrict__ m) {
  int idx = blockIdx.x * blockDim.x + threadIdx.x;
  if (idx >= ET * HH) return;
  int e = idx / HH, hd = idx - e * HH;
  int s, d;
  if (e < EE) { s = ei[e]; d = ei[EE + e]; } else { s = d = e - EE; }
  float v = es[s * HH + hd] + ed[d * HH + hd];
  v = (v >= 0.f) ? v : 0.2f * v;           // leaky_relu, PyG default slope
  e_buf[idx] = v;
  atomicMax(&m[d * HH + hd], f2ord(v));
}

// Pass 2: p = exp(e - m[dst]); segment-sum into z[dst].
template <int HH>
__global__ __launch_bounds__(256) void gat_edge_exp(const int* __restrict__ ei,
                                                    const float* __restrict__ e_buf,
                                                    const unsigned* __restrict__ m,
                                                    float* __restrict__ p_buf,
                                                    float* __restrict__ z) {
  int idx = blockIdx.x * blockDim.x + threadIdx.x;
  if (idx >= ET * HH) return;
  int e = idx / HH, hd = idx - e * HH;
  int d = (e < EE) ? ei[EE + e] : (e - EE);
  float p = expf(e_buf[idx] - ord2f(m[d * HH + hd]));
  p_buf[idx] = p;
  atomicAdd(&z[d * HH + hd], p);
}

// Pass 3: agg[dst,c] += (p/(z[dst]+eps)) * h[src,c].  Thread per (edge, channel).
template <int HH>
__global__ __launch_bounds__(256) void gat_edge_aggr(const int* __restrict__ ei,
                                                     const float* __restrict__ Hf,
                                                     const float* __restrict__ p_buf,
                                                     const float* __restrict__ z,
                                                     float* __restrict__ agg) {
  int idx = blockIdx.x * blockDim.x + threadIdx.x;
  if (idx >= ET * DD) return;
  int e = idx >> 7;                         // / DD
  int c = idx & (DD - 1);
  int hd = (HH == 1) ? 0 : (c >> 5);        // C = 128/HH
  int s, d;
  if (e < EE) { s = ei[e]; d = ei[EE + e]; } else { s = d = e - EE; }
  float alpha = p_buf[e * HH + hd] / (z[d * HH + hd] + 1e-16f);
  atomicAdd(&agg[d * DD + c], alpha * Hf[s * DD + c]);
}

// out = (agg + bias) [with optional relu]
__global__ __launch_bounds__(256) void gat_finish(const float* __restrict__ agg,
                                                  const float* __restrict__ b,
                                                  float* __restrict__ out,
                                                  int do_relu) {
  int idx = blockIdx.x * blockDim.x + threadIdx.x;
  if (idx >= NN * DD) return;
  float v = agg[idx] + b[idx & (DD - 1)];
  out[idx] = do_relu ? fmaxf(v, 0.f) : v;
}

// ---------------------------------------------------------------------------
extern "C" void kernel_launch(void* const* d_in, const int* in_sizes, int n_in,
                              void* d_out, int out_size, void* d_ws, size_t ws_size,
                              hipStream_t stream) {
  const float* x        = (const float*)d_in[0];
  const int*   ei       = (const int*)d_in[1];      // [2, E] int32
  const float* Ws       = (const float*)d_in[2];    // [19,128,128]
  const float* a_src    = (const float*)d_in[3];    // [19,4,32]
  const float* a_dst    = (const float*)d_in[4];    // [19,4,32]
  const float* bs       = (const float*)d_in[5];    // [19,128]
  const float* W_last   = (const float*)d_in[6];    // [128,128]
  const float* a_src_l  = (const float*)d_in[7];    // [1,128]
  const float* a_dst_l  = (const float*)d_in[8];    // [1,128]
  const float* b_last   = (const float*)d_in[9];    // [128]

  char* ws = (char*)d_ws;
  float*    h     = (float*)(ws + 0);           // 5,120,000 B
  float*    agg   = (float*)(ws + 5120000);     // 5,120,000 B
  float*    xb    = (float*)(ws + 10240000);    // 5,120,000 B
  float*    es    = (float*)(ws + 15360000);    //   160,000 B
  float*    ed    = (float*)(ws + 15520000);    //   160,000 B
  unsigned* m     = (unsigned*)(ws + 15680000); //   160,000 B
  float*    z     = (float*)(ws + 15840000);    //   160,000 B
  float*    e_buf = (float*)(ws + 16000000);    // 2,720,000 B
  float*    p_buf = (float*)(ws + 18720000);    // 2,720,000 B

  const dim3 blk(256);
  const int g_nd  = (NN * DD + 255) / 256;   // node*channel grids
  const int g_nh4 = (NN * 4 + 255) / 256;
  const int g_nh1 = (NN + 255) / 256;
  const int g_eh4 = (ET * 4 + 255) / 256;
  const int g_eh1 = (ET + 255) / 256;
  const int g_ed  = (ET * DD + 255) / 256;

  const float* cur = x;
  for (int l = 0; l < LMID; ++l) {
    const float* W  = Ws + (size_t)l * DD * DD;
    const float* as = a_src + (size_t)l * 4 * 32;
    const float* ad = a_dst + (size_t)l * 4 * 32;
    const float* b  = bs + (size_t)l * DD;

    gat_gemm<<<625, blk, 0, stream>>>(cur, W, h);
    gat_attn<4, 32><<<g_nh4, blk, 0, stream>>>(h, as, ad, es, ed);
    gat_init<<<g_nd, blk, 0, stream>>>(m, z, agg, NN * 4);
    gat_edge_max<4><<<g_eh4, blk, 0, stream>>>(ei, es, ed, e_buf, m);
    gat_edge_exp<4><<<g_eh4, blk, 0, stream>>>(ei, e_buf, m, p_buf, z);
    gat_edge_aggr<4><<<g_ed, blk, 0, stream>>>(ei, h, p_buf, z, agg);
    gat_finish<<<g_nd, blk, 0, stream>>>(agg, b, xb, /*relu=*/1);
    cur = xb;
  }

  // Final layer: heads=1, C=128, no relu, mean over 1 head == identity.
  gat_gemm<<<625, blk, 0, stream>>>(cur, W_last, h);
  gat_attn<1, 128><<<g_nh1, blk, 0, stream>>>(h, a_src_l, a_dst_l, es, ed);
  gat_init<<<g_nd, blk, 0, stream>>>(m, z, agg, NN);
  gat_edge_max<1><<<g_eh1, blk, 0, stream>>>(ei, es, ed, e_buf, m);
  gat_edge_exp<1><<<g_eh1, blk, 0, stream>>>(ei, e_buf, m, p_buf, z);
  gat_edge_aggr<1><<<g_ed, blk, 0, stream>>>(ei, h, p_buf, z, agg);
  gat_finish<<<g_nd, blk, 0, stream>>>(agg, b_last, (float*)d_out, /*relu=*/0);
}